// Model_74431783240179
// MI455X (gfx1250) — compile-verified
//
#include <hip/hip_runtime.h>

#define B_    32
#define N_    300
#define T_    288
#define FIN   10
#define H_    64
#define E_    2400
#define PRED  144
#define CIN   74            // FIN + H
#define ROWS  (B_*N_)       // 9600
#define G3    192           // 3*H

// padded LDS row strides (row-to-row offset == 32 mod 64 banks for the
// two half-wave K-rows of a B fragment -> conflict-free ds_loads)
#define WC_S  80            // W_conv rows (64 cols)
#define WG_S  208           // W_ih / W_hh rows (192 cols)

typedef float v2f __attribute__((ext_vector_type(2)));
typedef float v8f __attribute__((ext_vector_type(8)));

__device__ __forceinline__ float sigm_(float x) { return 1.0f / (1.0f + __expf(-x)); }

// ---------------------------------------------------------------------------
// One-time deterministic CSR build (dst-grouped, original edge order kept so
// the gather summation order is bit-stable across replays).
// edge_index flat layout: [0..E) = src, [E..2E) = dst
// ---------------------------------------------------------------------------
__global__ void build_csr(const int* __restrict__ edge,
                          int* __restrict__ off, int* __restrict__ srcs,
                          int* __restrict__ cnt) {
    if (blockIdx.x != 0 || threadIdx.x != 0) return;
    for (int n = 0; n < N_; ++n) cnt[n] = 0;
    for (int e = 0; e < E_; ++e) cnt[edge[E_ + e]]++;
    off[0] = 0;
    for (int n = 0; n < N_; ++n) off[n + 1] = off[n] + cnt[n];
    for (int n = 0; n < N_; ++n) cnt[n] = 0;
    for (int e = 0; e < E_; ++e) {
        int d = edge[E_ + e];
        srcs[off[d] + cnt[d]] = edge[e];
        cnt[d]++;
    }
}

__global__ void zero_f(float* __restrict__ p, int n) {
    int i = blockIdx.x * blockDim.x + threadIdx.x;
    for (; i < n; i += gridDim.x * blockDim.x) p[i] = 0.0f;
}

// ---------------------------------------------------------------------------
// Diffusion hop 1, fused with cat = [x, h] construction.
// out[b,n,c] = cat(b,n,c) + sum_{e: dst=n} cat(b,src_e,c)
// ---------------------------------------------------------------------------
__global__ void hop_cat(const float* __restrict__ x, int xstride,
                        const float* __restrict__ h,
                        float* __restrict__ out,
                        const int* __restrict__ off, const int* __restrict__ srcs) {
    int bn = blockIdx.x;            // 0..ROWS-1
    int c  = threadIdx.x;
    if (c >= CIN) return;
    int b    = bn / N_;
    int n    = bn - b * N_;
    int base = b * N_;
    float acc = (c < FIN) ? x[(size_t)(base + n) * xstride + c]
                          : h[(size_t)(base + n) * H_ + (c - FIN)];
    int e1 = off[n + 1];
    for (int e = off[n]; e < e1; ++e) {
        int s = srcs[e];
        acc += (c < FIN) ? x[(size_t)(base + s) * xstride + c]
                         : h[(size_t)(base + s) * H_ + (c - FIN)];
    }
    out[(size_t)bn * CIN + c] = acc;
}

// Diffusion hop 2 on a dense [ROWS, CIN] buffer.
__global__ void hop_gen(const float* __restrict__ in, float* __restrict__ out,
                        const int* __restrict__ off, const int* __restrict__ srcs) {
    int bn = blockIdx.x;
    int c  = threadIdx.x;
    if (c >= CIN) return;
    int b    = bn / N_;
    int n    = bn - b * N_;
    int base = b * N_;
    float acc = in[(size_t)bn * CIN + c];
    int e1 = off[n + 1];
    for (int e = off[n]; e < e1; ++e)
        acc += in[(size_t)(base + srcs[e]) * CIN + c];
    out[(size_t)bn * CIN + c] = acc;
}

// ---------------------------------------------------------------------------
// Fused conv-GEMM + GRU cell. One wave per 16-row M tile; fp32 WMMA 16x16x4.
// All weights staged in LDS once per block (coalesced), so every B fragment
// in the WMMA K-loops is a conflict-free ds_load instead of a scattered
// global_load_b32.
// ---------------------------------------------------------------------------
__global__ __launch_bounds__(256)
void cell_kernel(const float* __restrict__ xc,           // [ROWS, CIN]
                 const float* __restrict__ x, int xstride,
                 float* __restrict__ h_out,              // [ROWS, H]
                 const float* __restrict__ Wc,  const float* __restrict__ bc,
                 const float* __restrict__ Wih, const float* __restrict__ bih,
                 const float* __restrict__ Whh, const float* __restrict__ bhh) {
    __shared__ float sWc [CIN * WC_S];          // 74 x 80
    __shared__ float sWih[FIN * WG_S];          // 10 x 208
    __shared__ float sWhh[H_  * WG_S];          // 64 x 208
    __shared__ float sB  [H_ + 2 * G3];         // bc | bih | bhh
    __shared__ float convs[8][16][H_ + 4];

    int tid = threadIdx.x;
    for (int i = tid; i < CIN * H_; i += 256) {
        int r = i / H_;
        sWc[r * WC_S + (i - r * H_)] = Wc[i];
    }
    for (int i = tid; i < FIN * G3; i += 256) {
        int r = i / G3;
        sWih[r * WG_S + (i - r * G3)] = Wih[i];
    }
    for (int i = tid; i < H_ * G3; i += 256) {
        int r = i / G3;
        sWhh[r * WG_S + (i - r * G3)] = Whh[i];
    }
    for (int i = tid; i < H_; i += 256) sB[i] = bc[i];
    for (int i = tid; i < G3; i += 256) sB[H_ + i] = bih[i];
    for (int i = tid; i < G3; i += 256) sB[H_ + G3 + i] = bhh[i];
    __syncthreads();

    int wave  = tid >> 5;
    int lane  = tid & 31;
    int rbase = (blockIdx.x * 8 + wave) * 16;
    int m     = lane & 15;          // A-frag row / LDS row
    int kb    = (lane >> 4) << 1;   // K sub-offset: 0 or 2
    int nn    = lane & 15;          // B/D-frag column
    int hi    = lane >> 4;          // D-frag M-half

    // ---- conv_out tile: [16 x 64] = xc[16 x 74] @ Wc[74 x 64] ------------
    const float* xrow = xc + (size_t)(rbase + m) * CIN;
    v8f acc0 = {}; v8f acc1 = {}; v8f acc2 = {}; v8f acc3 = {};
    for (int kk = 0; kk < 19; ++kk) {           // 19*4 = 76 >= 74
        int k0 = kk * 4 + kb;
        v2f a, b0, b1, b2, b3;
        a.x = (k0     < CIN) ? xrow[k0]     : 0.0f;
        a.y = (k0 + 1 < CIN) ? xrow[k0 + 1] : 0.0f;
        if (k0 < CIN) {
            const float* w = sWc + k0 * WC_S;
            b0.x = w[nn]; b1.x = w[16 + nn]; b2.x = w[32 + nn]; b3.x = w[48 + nn];
        } else { b0.x = b1.x = b2.x = b3.x = 0.0f; }
        if (k0 + 1 < CIN) {
            const float* w = sWc + (k0 + 1) * WC_S;
            b0.y = w[nn]; b1.y = w[16 + nn]; b2.y = w[32 + nn]; b3.y = w[48 + nn];
        } else { b0.y = b1.y = b2.y = b3.y = 0.0f; }
        acc0 = __builtin_amdgcn_wmma_f32_16x16x4_f32(false, a, false, b0, (short)0, acc0, false, false);
        acc1 = __builtin_amdgcn_wmma_f32_16x16x4_f32(false, a, false, b1, (short)0, acc1, false, false);
        acc2 = __builtin_amdgcn_wmma_f32_16x16x4_f32(false, a, false, b2, (short)0, acc2, false, false);
        acc3 = __builtin_amdgcn_wmma_f32_16x16x4_f32(false, a, false, b3, (short)0, acc3, false, false);
    }
    #pragma unroll
    for (int r = 0; r < 8; ++r) {
        int mr = hi * 8 + r;
        convs[wave][mr][ 0 + nn] = acc0[r] + sB[ 0 + nn];
        convs[wave][mr][16 + nn] = acc1[r] + sB[16 + nn];
        convs[wave][mr][32 + nn] = acc2[r] + sB[32 + nn];
        convs[wave][mr][48 + nn] = acc3[r] + sB[48 + nn];
    }
    __syncthreads();

    // ---- GRU gates -------------------------------------------------------
    const float* xr = x + (size_t)(rbase + m) * xstride;
    v2f xa[3];
    #pragma unroll
    for (int kk = 0; kk < 3; ++kk) {            // K=10 padded to 12
        int k0 = kk * 4 + kb;
        xa[kk].x = (k0     < FIN) ? xr[k0]     : 0.0f;
        xa[kk].y = (k0 + 1 < FIN) ? xr[k0 + 1] : 0.0f;
    }

    for (int j = 0; j < 4; ++j) {               // 4 column tiles over H=64
        int col = j * 16 + nn;
        v8f gir = {}; v8f giz = {}; v8f gin = {};
        v8f ghr = {}; v8f ghz = {}; v8f ghn = {};

        // gi = x @ Wih  (r/z/n gate column blocks share A)
        #pragma unroll
        for (int kk = 0; kk < 3; ++kk) {
            int k0 = kk * 4 + kb;
            v2f br, bz, bn2;
            if (k0 < FIN) {
                const float* w = sWih + k0 * WG_S;
                br.x = w[col]; bz.x = w[H_ + col]; bn2.x = w[2 * H_ + col];
            } else { br.x = bz.x = bn2.x = 0.0f; }
            if (k0 + 1 < FIN) {
                const float* w = sWih + (k0 + 1) * WG_S;
                br.y = w[col]; bz.y = w[H_ + col]; bn2.y = w[2 * H_ + col];
            } else { br.y = bz.y = bn2.y = 0.0f; }
            gir = __builtin_amdgcn_wmma_f32_16x16x4_f32(false, xa[kk], false, br,  (short)0, gir, false, false);
            giz = __builtin_amdgcn_wmma_f32_16x16x4_f32(false, xa[kk], false, bz,  (short)0, giz, false, false);
            gin = __builtin_amdgcn_wmma_f32_16x16x4_f32(false, xa[kk], false, bn2, (short)0, gin, false, false);
        }
        // gh = conv_out @ Whh  (A from LDS conv tile, K=64)
        #pragma unroll
        for (int kk = 0; kk < 16; ++kk) {
            int k0 = kk * 4 + kb;
            v2f a, br, bz, bn2;
            a.x = convs[wave][m][k0];
            a.y = convs[wave][m][k0 + 1];
            const float* w0 = sWhh + k0 * WG_S;
            const float* w1 = w0 + WG_S;
            br.x  = w0[col];           br.y  = w1[col];
            bz.x  = w0[H_ + col];      bz.y  = w1[H_ + col];
            bn2.x = w0[2 * H_ + col];  bn2.y = w1[2 * H_ + col];
            ghr = __builtin_amdgcn_wmma_f32_16x16x4_f32(false, a, false, br,  (short)0, ghr, false, false);
            ghz = __builtin_amdgcn_wmma_f32_16x16x4_f32(false, a, false, bz,  (short)0, ghz, false, false);
            ghn = __builtin_amdgcn_wmma_f32_16x16x4_f32(false, a, false, bn2, (short)0, ghn, false, false);
        }
        // gates + output
        #pragma unroll
        for (int r = 0; r < 8; ++r) {
            int mr = hi * 8 + r;
            float ir  = gir[r] + sB[H_ + col];
            float iz  = giz[r] + sB[H_ + H_ + col];
            float in_ = gin[r] + sB[H_ + 2 * H_ + col];
            float hr  = ghr[r] + sB[H_ + G3 + col];
            float hz  = ghz[r] + sB[H_ + G3 + H_ + col];
            float hn  = ghn[r] + sB[H_ + G3 + 2 * H_ + col];
            float co  = convs[wave][mr][col];
            float rg  = sigm_(ir + hr);
            float zg  = sigm_(iz + hz);
            float ng  = tanhf(in_ + rg * hn);
            h_out[(size_t)(rbase + mr) * H_ + col] = (1.0f - zg) * ng + zg * co;
        }
    }
}

// ---------------------------------------------------------------------------
// Decoder readout: out = h @ W_out + b_out (OUT=1); dec_in = out @ W_o2i + b.
// ---------------------------------------------------------------------------
__global__ void dec_kernel(const float* __restrict__ h,
                           const float* __restrict__ Wo,  const float* __restrict__ bo,
                           const float* __restrict__ Wo2i, const float* __restrict__ bo2i,
                           float* __restrict__ y, float* __restrict__ dec_in, int step) {
    int row = blockIdx.x * blockDim.x + threadIdx.x;
    if (row >= ROWS) return;
    const float* hr = h + (size_t)row * H_;
    float acc = bo[0];
    #pragma unroll
    for (int c = 0; c < H_; ++c) acc += hr[c] * Wo[c];
    y[(size_t)row * PRED + step] = acc;
    #pragma unroll
    for (int f = 0; f < FIN; ++f)
        dec_in[(size_t)row * FIN + f] = acc * Wo2i[f] + bo2i[f];
}

// ---------------------------------------------------------------------------
extern "C" void kernel_launch(void* const* d_in, const int* in_sizes, int n_in,
                              void* d_out, int out_size, void* d_ws, size_t ws_size,
                              hipStream_t stream) {
    (void)in_sizes; (void)n_in; (void)out_size; (void)ws_size;
    const float* batch_x = (const float*)d_in[0];
    const int*   edge    = (const int*)d_in[1];
    const float* Wc      = (const float*)d_in[2];
    const float* bc      = (const float*)d_in[3];
    const float* Wih     = (const float*)d_in[4];
    const float* bih     = (const float*)d_in[5];
    const float* Whh     = (const float*)d_in[6];
    const float* bhh     = (const float*)d_in[7];
    const float* Wo      = (const float*)d_in[8];
    const float* bo      = (const float*)d_in[9];
    const float* Wo2i    = (const float*)d_in[10];
    const float* bo2i    = (const float*)d_in[11];
    float* y = (float*)d_out;

    char* ws = (char*)d_ws;
    auto carve = [&](size_t bytes) -> void* {
        void* p = (void*)ws;
        ws += (bytes + 255) & ~(size_t)255;
        return p;
    };
    float* h      = (float*)carve((size_t)ROWS * H_  * 4);
    float* bufA   = (float*)carve((size_t)ROWS * CIN * 4);
    float* bufB   = (float*)carve((size_t)ROWS * CIN * 4);
    float* dec_in = (float*)carve((size_t)ROWS * FIN * 4);
    int*   off    = (int*)carve((N_ + 1) * 4);
    int*   srcs   = (int*)carve(E_ * 4);
    int*   cnt    = (int*)carve(N_ * 4);

    build_csr<<<1, 1, 0, stream>>>(edge, off, srcs, cnt);
    zero_f<<<256, 256, 0, stream>>>(h, ROWS * H_);
    zero_f<<<64, 256, 0, stream>>>(dec_in, ROWS * FIN);

    // Encoder: 288 sequential cells
    for (int t = 0; t < T_; ++t) {
        const float* xt = batch_x + (size_t)t * FIN;   // row stride T*FIN
        hop_cat<<<ROWS, 96, 0, stream>>>(xt, T_ * FIN, h, bufB, off, srcs);
        hop_gen<<<ROWS, 96, 0, stream>>>(bufB, bufA, off, srcs);
        cell_kernel<<<ROWS / 128, 256, 0, stream>>>(bufA, xt, T_ * FIN, h,
                                                    Wc, bc, Wih, bih, Whh, bhh);
    }
    // Decoder: 144 sequential cells + readout
    for (int s = 0; s < PRED; ++s) {
        hop_cat<<<ROWS, 96, 0, stream>>>(dec_in, FIN, h, bufB, off, srcs);
        hop_gen<<<ROWS, 96, 0, stream>>>(bufB, bufA, off, srcs);
        cell_kernel<<<ROWS / 128, 256, 0, stream>>>(bufA, dec_in, FIN, h,
                                                    Wc, bc, Wih, bih, Whh, bhh);
        dec_kernel<<<(ROWS + 255) / 256, 256, 0, stream>>>(h, Wo, bo, Wo2i, bo2i,
                                                           y, dec_in, s);
    }
}